// RGCNGatherMMUnsorted_13099650253295
// MI455X (gfx1250) — compile-verified
//
#include <hip/hip_runtime.h>

typedef __attribute__((ext_vector_type(2))) float v2f;
typedef __attribute__((ext_vector_type(8))) float v8f;

#define RGCN_RELS   32
#define RGCN_DIN    64
#define RGCN_DOUT   64
#define RGCN_RCHUNK 8              // relations per wave (32/8 = 4 r-chunks)

// ---------------------------------------------------------------------------
// Zero-fill (harness poisons d_out / d_ws with 0xAA).
// ---------------------------------------------------------------------------
__global__ void rgcn_zero_f32(float* __restrict__ p, long long n) {
    long long i = (long long)blockIdx.x * blockDim.x + threadIdx.x;
    long long stride = (long long)gridDim.x * blockDim.x;
    for (; i < n; i += stride) p[i] = 0.0f;
}

// ---------------------------------------------------------------------------
// Phase 1: scatter-accumulate gathered source features into the per-(node,rel)
// accumulator: acc[(dst*R + et) * 64 + k] += feat[src*64 + k].
// 16 lanes per edge, 4 floats (one b128 load) per lane, f32 global atomics.
// ---------------------------------------------------------------------------
__global__ void rgcn_scatter_accum(const float* __restrict__ feat,
                                   const int*   __restrict__ src,
                                   const int*   __restrict__ dst,
                                   const int*   __restrict__ et,
                                   float*       __restrict__ acc,
                                   int n_edges) {
    long long idx = (long long)blockIdx.x * blockDim.x + threadIdx.x;
    int e = (int)(idx >> 4);
    int p = (int)(idx & 15);
    if (e >= n_edges) return;
    int s = src[e];
    int d = dst[e];
    int r = et[e];
    const float4 v = *(const float4*)(feat + (long long)s * RGCN_DIN + p * 4);
    float* a = acc + ((long long)d * RGCN_RELS + r) * RGCN_DIN + p * 4;
    atomicAdd(a + 0, v.x);
    atomicAdd(a + 1, v.y);
    atomicAdd(a + 2, v.z);
    atomicAdd(a + 3, v.w);
}

// ---------------------------------------------------------------------------
// Phase 2: out[n,o] += sum_{r in chunk} sum_k acc[n,r,k] * W[r,k,o]
//
// One wave32 per (16-node tile, r-chunk of 8 relations). The wave produces the
// full 16x64 output slab as four 16x16 f32 WMMA accumulators, so each A load
// (disjoint across waves -> acc read exactly once, 410 MB total) feeds 4 WMMAs.
// Partial results from the 4 r-chunks are combined via f32 global atomics.
//
// V_WMMA_F32_16X16X4_F32 operand layouts (ISA 7.12.2):
//  A (16x4):  lanes 0-15 row M=lane,    VGPR0=K0, VGPR1=K1;
//             lanes 16-31 row M=lane-16, VGPR0=K2, VGPR1=K3.
//  B (4x16):  VGPR j: lanes 0-15 K=j (N=lane), lanes 16-31 K=j+2 (N=lane-16).
//  C/D (16x16): VGPR i = row i (lanes 0-15) / row i+8 (lanes 16-31), N=lane%16.
// ---------------------------------------------------------------------------
__global__ void rgcn_gemm_wmma(const float* __restrict__ acc,
                               const float* __restrict__ weight,
                               float*       __restrict__ out,
                               int n_tiles_total) {
    const int lane = threadIdx.x & 31;
    const int wave = threadIdx.x >> 5;                 // 4 waves per block
    const int tile = blockIdx.x * 4 + wave;            // wave-uniform
    if (tile >= n_tiles_total) return;                 // uniform: EXEC all-1s inside

    const int nt  = tile >> 2;                         // node tile (16 nodes)
    const int rc  = tile & 3;                          // relation chunk (8 rels)
    const int row = lane & 15;                         // A row / B,C column
    const int hi  = lane >> 4;                         // K-half select (0 or 1)

    const float* arow_base =
        acc + ((long long)(nt * 16 + row)) * (RGCN_RELS * RGCN_DIN);
    const float* wcol_base = weight + row;             // + r*DIN*DOUT + k*DOUT + ot*16

    v8f c0 = {}, c1 = {}, c2 = {}, c3 = {};
    for (int rr = 0; rr < RGCN_RCHUNK; ++rr) {
        const int r = rc * RGCN_RCHUNK + rr;
        const float* arow = arow_base + (long long)r * RGCN_DIN;
        const float* wrel = wcol_base + (long long)r * RGCN_DIN * RGCN_DOUT;
#pragma unroll
        for (int k = 0; k < RGCN_DIN; k += 4) {
            // A: one 8-byte load feeds all four output-column tiles
            v2f a = *(const v2f*)(arow + k + hi * 2);
            const float* bk = wrel + (long long)(k + hi * 2) * RGCN_DOUT;

            v2f b0; b0.x = bk[0];  b0.y = bk[RGCN_DOUT + 0];
            v2f b1; b1.x = bk[16]; b1.y = bk[RGCN_DOUT + 16];
            v2f b2; b2.x = bk[32]; b2.y = bk[RGCN_DOUT + 32];
            v2f b3; b3.x = bk[48]; b3.y = bk[RGCN_DOUT + 48];

            // 8 args: (neg_a, A, neg_b, B, c_mod, C, reuse_a, reuse_b)
            c0 = __builtin_amdgcn_wmma_f32_16x16x4_f32(false, a, false, b0,
                                                       (short)0, c0, false, false);
            c1 = __builtin_amdgcn_wmma_f32_16x16x4_f32(false, a, false, b1,
                                                       (short)0, c1, false, false);
            c2 = __builtin_amdgcn_wmma_f32_16x16x4_f32(false, a, false, b2,
                                                       (short)0, c2, false, false);
            c3 = __builtin_amdgcn_wmma_f32_16x16x4_f32(false, a, false, b3,
                                                       (short)0, c3, false, false);
        }
    }

    // Combine this r-chunk's partial 16x64 slab into out with f32 atomics.
#pragma unroll
    for (int i = 0; i < 8; ++i) {
        const int m = i + hi * 8;                      // C/D row mapping
        float* orow = out + ((long long)(nt * 16 + m)) * RGCN_DOUT + row;
        atomicAdd(orow + 0,  c0[i]);
        atomicAdd(orow + 16, c1[i]);
        atomicAdd(orow + 32, c2[i]);
        atomicAdd(orow + 48, c3[i]);
    }
}

// ---------------------------------------------------------------------------
// Fallback if d_ws cannot hold the [N, R, 64] f32 accumulator: direct per-edge
// matvec + atomic scatter into out. One thread per (edge, output column).
// ---------------------------------------------------------------------------
__global__ void rgcn_direct(const float* __restrict__ feat,
                            const float* __restrict__ weight,
                            const int*   __restrict__ src,
                            const int*   __restrict__ dst,
                            const int*   __restrict__ et,
                            float*       __restrict__ out,
                            int n_edges) {
    long long idx = (long long)blockIdx.x * blockDim.x + threadIdx.x;
    int e = (int)(idx >> 6);
    int o = (int)(idx & 63);
    if (e >= n_edges) return;
    const float* f = feat + (long long)src[e] * RGCN_DIN;
    const float* w = weight + (long long)et[e] * RGCN_DIN * RGCN_DOUT + o;
    float s = 0.0f;
#pragma unroll
    for (int k = 0; k < RGCN_DIN; ++k) s = fmaf(f[k], w[k * RGCN_DOUT], s);
    atomicAdd(out + (long long)dst[e] * RGCN_DOUT + o, s);
}

// ---------------------------------------------------------------------------
// Launch: inputs are (feat, weight, src, dst, etypes, E_per_rel) per reference.
// ---------------------------------------------------------------------------
extern "C" void kernel_launch(void* const* d_in, const int* in_sizes, int n_in,
                              void* d_out, int out_size, void* d_ws, size_t ws_size,
                              hipStream_t stream) {
    const float* feat   = (const float*)d_in[0];
    const float* weight = (const float*)d_in[1];
    const int*   src    = (const int*)d_in[2];
    const int*   dst    = (const int*)d_in[3];
    const int*   et     = (const int*)d_in[4];
    float*       out    = (float*)d_out;

    const int n_edges = in_sizes[2];
    const int n_nodes = in_sizes[0] / RGCN_DIN;

    const size_t acc_elems = (size_t)n_nodes * RGCN_RELS * RGCN_DIN;
    const size_t acc_bytes = acc_elems * sizeof(float);

    // zero the output (harness poisons it; GEMM phase accumulates atomically)
    rgcn_zero_f32<<<1024, 256, 0, stream>>>(out, (long long)n_nodes * RGCN_DOUT);

    if (ws_size >= acc_bytes && (n_nodes % 16) == 0) {
        float* acc = (float*)d_ws;
        rgcn_zero_f32<<<8192, 256, 0, stream>>>(acc, (long long)acc_elems);

        long long scat_work = (long long)n_edges * 16;
        int scat_blocks = (int)((scat_work + 255) / 256);
        rgcn_scatter_accum<<<scat_blocks, 256, 0, stream>>>(feat, src, dst, et, acc, n_edges);

        // one wave per (node tile, r-chunk): 3125 * 4 = 12500 waves
        int n_tiles = (n_nodes / 16) * (RGCN_RELS / RGCN_RCHUNK);
        int blocks  = (n_tiles + 3) / 4;               // 4 waves (128 threads) per block
        rgcn_gemm_wmma<<<blocks, 128, 0, stream>>>(acc, weight, out, n_tiles);
    } else {
        long long work = (long long)n_edges * 64;
        int blocks = (int)((work + 255) / 256);
        rgcn_direct<<<blocks, 256, 0, stream>>>(feat, weight, src, dst, et, out, n_edges);
    }
}